// SpatialGroupingModule_5935644803758
// MI455X (gfx1250) — compile-verified
//
#include <hip/hip_runtime.h>
#include <hip/hip_bf16.h>

typedef float v2f __attribute__((ext_vector_type(2)));
typedef float v8f __attribute__((ext_vector_type(8)));

#define WMMA_F32X4(A, B, C) \
  __builtin_amdgcn_wmma_f32_16x16x4_f32(false, (A), false, (B), (short)0, (C), false, false)

#define N_NODES 8192
#define T_STEPS 12

// ---------------------------------------------------------------------------
// 3-tap conv along N as implicit GEMM, fp32 WMMA 16x16x4.
// Block tile: 64(M) x 128(N), 8 waves; wave = one 16-row M-subtile x four
// 16-col N tiles. Activations staged in LDS with halo; weights read via L2.
// xmode=1: input is x[B,T,N,64] with channel c = t*64+d (im2row of h).
// ---------------------------------------------------------------------------
__global__ __launch_bounds__(256) void conv3x1_gemm(
    const float* __restrict__ Xin, const float* __restrict__ W,
    const float* __restrict__ bias, float* __restrict__ Y,
    int C, int Mtot, int relu, int xmode)
{
  __shared__ float xs[16][132];
  const int b  = blockIdx.z;
  const int n0 = blockIdx.x * 128;
  const int m0 = blockIdx.y * 64;
  const int tid = threadIdx.x;
  const int lane = tid & 31, wid = tid >> 5;
  const int msub = wid & 3, nh = wid >> 2;   // nh in 0..1
  const int mo = m0 + msub * 16;
  const int lr = lane & 15, lh = lane >> 4;
  v8f acc[4] = {};

  for (int c0 = 0; c0 < C; c0 += 16) {
    __syncthreads();
    for (int e = tid; e < 16 * 130; e += 256) {
      int cc = e / 130, nn = e % 130;
      int n = n0 + nn - 1;
      float v = 0.f;
      if (n >= 0 && n < N_NODES) {
        int c = c0 + cc;
        if (xmode)
          v = Xin[(((b * T_STEPS + (c >> 6)) * N_NODES + n) << 6) + (c & 63)];
        else
          v = Xin[((size_t)b * C + c) * N_NODES + n];
      }
      xs[cc][nn] = v;
    }
    __syncthreads();
    #pragma unroll
    for (int r = 0; r < 3; ++r) {
      #pragma unroll
      for (int kk = 0; kk < 4; ++kk) {
        const int cA = c0 + kk * 4 + lh * 2;
        // weight element (o, c, r, kw=1) of [O][C][3][3]
        const float* wp = W + ((size_t)(mo + lr) * C + cA) * 9 + r * 3 + 1;
        v2f a; a.x = wp[0]; a.y = wp[9];
        const int kb = kk * 4 + lh * 2;
        #pragma unroll
        for (int tt = 0; tt < 4; ++tt) {
          const int col = nh * 64 + tt * 16 + lr + r;
          v2f bf; bf.x = xs[kb][col]; bf.y = xs[kb + 1][col];
          acc[tt] = WMMA_F32X4(a, bf, acc[tt]);
        }
      }
    }
  }
  #pragma unroll
  for (int tt = 0; tt < 4; ++tt) {
    #pragma unroll
    for (int j = 0; j < 8; ++j) {
      const int o = mo + j + lh * 8;
      const int n = n0 + nh * 64 + tt * 16 + lr;
      float v = acc[tt][j] + bias[o];
      if (relu) v = fmaxf(v, 0.f);
      Y[((size_t)b * Mtot + o) * N_NODES + n] = v;
    }
  }
}

// ---------------------------------------------------------------------------
// Fused: wc conv2 (64->2048, 3-tap) + row softmax over [512, 4*N] + c = P@zs.
// Block = (a', b): 16 conv channels = 4 softmax rows, ALL 8192 nodes ->
// softmax rows are block-local; online (flash) softmax, wcf never hits HBM.
// w1/t re-reads across the 128 a'-blocks hit L2 (4.2MB/batch << 192MB).
// ---------------------------------------------------------------------------
__global__ __launch_bounds__(128) void fused_wcf_softmax_c(
    const float* __restrict__ w1, const float* __restrict__ W2,
    const float* __restrict__ b2, const float* __restrict__ tbuf,
    float* __restrict__ cout)
{
  __shared__ float wl[64 * 3 * 16];   // weights for this block's 16 channels
  __shared__ float xs[64][68];        // w1 tile with halo
  __shared__ float zl[64][68];        // zs tile (padded: conflict-free d-reads)
  __shared__ float praw[16][68];      // raw logits -> exp values
  __shared__ float Mx[4], Ssum[4], fup[4], stepred[4];
  __shared__ float V[4][16];

  const int ap = blockIdx.x;          // a' in 0..127
  const int b  = blockIdx.y;
  const int o0 = ap * 16;
  const int tid = threadIdx.x, lane = tid & 31, wid = tid >> 5;
  const int lr = lane & 15, lh = lane >> 4;

  for (int e = tid; e < 64 * 3 * 16; e += 128) {
    int oo = e % 16, q = e / 16;
    int rr = q % 3, cc = q / 3;
    wl[e] = W2[((size_t)(o0 + oo) * 64 + cc) * 9 + rr * 3 + 1];
  }
  if (tid < 4) { Mx[tid] = -3.0e38f; Ssum[tid] = 0.f; }
  if (tid < 64) V[tid >> 4][tid & 15] = 0.f;

  for (int n0 = 0; n0 < N_NODES; n0 += 64) {
    __syncthreads();
    for (int e = tid; e < 64 * 66; e += 128) {
      int cc = e / 66, nn = e % 66;
      int n = n0 + nn - 1;
      xs[cc][nn] = (n >= 0 && n < N_NODES)
                 ? w1[((size_t)b * 64 + cc) * N_NODES + n] : 0.f;
    }
    for (int e = tid; e < 64 * 64; e += 128) {
      int dd = e >> 6, col = e & 63;
      int g = col >> 4, i = col & 15;   // zs[m=g*8192+n] = t[(n&3)*16+d][g*2048+n/4]
      zl[dd][col] = tbuf[((size_t)b * 64 + dd) * N_NODES + (g * 2048 + (n0 >> 2) + i)];
    }
    __syncthreads();

    v8f acc = {};   // wave `wid` owns node sub-tile [n0+16*wid, +16)
    #pragma unroll
    for (int r = 0; r < 3; ++r) {
      for (int c4 = 0; c4 < 16; ++c4) {
        const int cA = c4 * 4 + lh * 2;
        v2f a; a.x = wl[(cA * 3 + r) * 16 + lr];
               a.y = wl[((cA + 1) * 3 + r) * 16 + lr];
        const int col = wid * 16 + lr + r;
        v2f bf; bf.x = xs[cA][col]; bf.y = xs[cA + 1][col];
        acc = WMMA_F32X4(a, bf, acc);
      }
    }
    #pragma unroll
    for (int j = 0; j < 8; ++j) {
      const int ch = j + lh * 8;
      praw[ch][wid * 16 + lr] = acc[j] + b2[o0 + ch];
    }
    __syncthreads();
    // step max: wave j reduces row j (channels 4j..4j+3, 64 nodes)
    float mloc = -3.0e38f;
    #pragma unroll
    for (int q = 0; q < 8; ++q) {
      int idx = lane + q * 32;
      mloc = fmaxf(mloc, praw[4 * wid + (idx >> 6)][idx & 63]);
    }
    for (int off = 16; off; off >>= 1) mloc = fmaxf(mloc, __shfl_xor(mloc, off, 32));
    if (lane == 0) stepred[wid] = mloc;
    __syncthreads();
    if (tid < 4) {
      float nm = fmaxf(Mx[tid], stepred[tid]);
      fup[tid] = __expf(Mx[tid] - nm);
      Mx[tid] = nm;
    }
    __syncthreads();
    for (int e = tid; e < 16 * 64; e += 128) {
      int ch = e >> 6;
      praw[ch][e & 63] = __expf(praw[ch][e & 63] - Mx[ch >> 2]);
    }
    __syncthreads();
    float sloc = 0.f;
    #pragma unroll
    for (int q = 0; q < 8; ++q) {
      int idx = lane + q * 32;
      sloc += praw[4 * wid + (idx >> 6)][idx & 63];
    }
    for (int off = 16; off; off >>= 1) sloc += __shfl_xor(sloc, off, 32);
    if (lane == 0) stepred[wid] = sloc;
    if (tid < 64) {               // V[j][d] = V*f + sum_{g,n} p * zs
      const int j = tid >> 4, d = tid & 15;
      float v = V[j][d] * fup[j];
      #pragma unroll 4
      for (int g = 0; g < 4; ++g)
        for (int n = 0; n < 64; ++n)
          v += praw[4 * j + g][n] * zl[(n & 3) * 16 + d][g * 16 + (n >> 2)];
      V[j][d] = v;
    }
    __syncthreads();
    if (tid < 4) Ssum[tid] = Ssum[tid] * fup[tid] + stepred[tid];
  }
  __syncthreads();
  if (tid < 64) {
    const int j = tid >> 4, d = tid & 15;
    cout[((size_t)b * 512 + (ap * 4 + j)) * 16 + d] = V[j][d] / Ssum[j];
  }
}

// ---------------------------------------------------------------------------
// dist = cdist(zs, c); ws_sm = softmax(-dist/TAU) over 512 clusters.
// Block: 512 threads (1 thread = 1 cluster), 64 zs-rows; c staged in LDS.
// ---------------------------------------------------------------------------
__global__ __launch_bounds__(512) void dist_softmax_ws(
    const float* __restrict__ tbuf, const float* __restrict__ cbuf,
    float* __restrict__ wsout)
{
  __shared__ float cl[512][17];
  __shared__ float zr[64][17];
  __shared__ float redA[16];
  __shared__ float redB[2];

  const int b  = blockIdx.y;
  const int m0 = blockIdx.x * 64;
  const int tid = threadIdx.x;
  const int lane = tid & 31, wv = tid >> 5;

  for (int e = tid; e < 512 * 16; e += 512)
    cl[e >> 4][e & 15] = cbuf[(size_t)b * 8192 + e];
  for (int e = tid; e < 64 * 16; e += 512) {
    int i = e >> 4, d = e & 15;
    int m = m0 + i;
    zr[i][d] = tbuf[((size_t)b * 64 + ((m & 3) * 16 + d)) * N_NODES + (m >> 2)];
  }
  __syncthreads();
  {
    float s = 0.f;
    #pragma unroll
    for (int d = 0; d < 16; ++d) { float v = cl[tid][d]; s += v * v; }
    cl[tid][16] = s;
  }
  if (tid < 64) {
    float s = 0.f;
    #pragma unroll
    for (int d = 0; d < 16; ++d) { float v = zr[tid][d]; s += v * v; }
    zr[tid][16] = s;
  }
  __syncthreads();

  for (int i = 0; i < 64; ++i) {
    float dot = 0.f;
    #pragma unroll
    for (int d = 0; d < 16; ++d) dot += zr[i][d] * cl[tid][d];
    float sq = zr[i][16] + cl[tid][16] - 2.f * dot;
    float logit = -sqrtf(fmaxf(sq, 1e-12f));   // TAU == 1.0
    float ml = logit;
    for (int off = 16; off; off >>= 1) ml = fmaxf(ml, __shfl_xor(ml, off, 32));
    if (lane == 0) redA[wv] = ml;
    __syncthreads();
    if (tid == 0) {
      float m = redA[0];
      for (int w = 1; w < 16; ++w) m = fmaxf(m, redA[w]);
      redB[0] = m;
    }
    __syncthreads();
    float p = __expf(logit - redB[0]);
    float sl = p;
    for (int off = 16; off; off >>= 1) sl += __shfl_xor(sl, off, 32);
    if (lane == 0) redA[wv] = sl;
    __syncthreads();
    if (tid == 0) {
      float s = 0.f;
      for (int w = 0; w < 16; ++w) s += redA[w];
      redB[1] = s;
    }
    __syncthreads();
    wsout[((size_t)b * 32768 + (m0 + i)) * 512 + tid] = p / redB[1];
  }
}

// ---------------------------------------------------------------------------
// Deterministic column sums (axis=1) + Y = ws_sm^T @ zs, chunked partials.
// ---------------------------------------------------------------------------
__global__ __launch_bounds__(512) void ws_colsum_part(
    const float* __restrict__ wsin, const float* __restrict__ tbuf,
    float* __restrict__ pcs, float* __restrict__ py)
{
  __shared__ float zr[512][16];
  const int b = blockIdx.y, ch = blockIdx.x;   // 64 chunks of 512 rows
  const int tid = threadIdx.x;                 // tid = cluster k
  const int mbase = ch * 512;
  for (int e = tid; e < 512 * 16; e += 512) {
    int i = e >> 4, d = e & 15;
    int m = mbase + i;
    zr[i][d] = tbuf[((size_t)b * 64 + ((m & 3) * 16 + d)) * N_NODES + (m >> 2)];
  }
  __syncthreads();
  float cs = 0.f;
  float yl[16];
  #pragma unroll
  for (int d = 0; d < 16; ++d) yl[d] = 0.f;
  for (int i = 0; i < 512; ++i) {
    float p = wsin[((size_t)b * 32768 + (mbase + i)) * 512 + tid];
    cs += p;
    #pragma unroll
    for (int d = 0; d < 16; ++d) yl[d] += p * zr[i][d];
  }
  pcs[((size_t)b * 64 + ch) * 512 + tid] = cs;
  #pragma unroll
  for (int d = 0; d < 16; ++d)
    py[(((size_t)b * 64 + ch) * 512 + tid) * 16 + d] = yl[d];
}

__global__ __launch_bounds__(512) void ws_colsum_reduce(
    const float* __restrict__ pcs, const float* __restrict__ py,
    float* __restrict__ colsumf, float* __restrict__ zc)
{
  const int b = blockIdx.x, tid = threadIdx.x;
  float cs = 0.f;
  for (int ch = 0; ch < 64; ++ch) cs += pcs[((size_t)b * 64 + ch) * 512 + tid];
  const float denom = cs + 1e-6f;
  colsumf[b * 512 + tid] = denom;
  const float inv = 1.f / denom;
  #pragma unroll
  for (int d = 0; d < 16; ++d) {
    float y = 0.f;
    for (int ch = 0; ch < 64; ++ch)
      y += py[(((size_t)b * 64 + ch) * 512 + tid) * 16 + d];
    zc[((size_t)b * 512 + tid) * 16 + d] = y * inv;
  }
}

__global__ __launch_bounds__(256) void ws_normalize(
    float* __restrict__ ws, const float* __restrict__ colsumf)
{
  const size_t idx = (size_t)blockIdx.x * 256 + threadIdx.x;  // float4 index
  const size_t e0 = idx * 4;
  const int k = (int)(e0 & 511);
  const int b = (int)(e0 >> 24);   // 32768*512 = 2^24 elements per batch
  float4 v = reinterpret_cast<float4*>(ws)[idx];
  const float* cf = colsumf + b * 512 + k;
  v.x /= cf[0]; v.y /= cf[1]; v.z /= cf[2]; v.w /= cf[3];
  reinterpret_cast<float4*>(ws)[idx] = v;
}

// ---------------------------------------------------------------------------
extern "C" void kernel_launch(void* const* d_in, const int* in_sizes, int n_in,
                              void* d_out, int out_size, void* d_ws, size_t ws_size,
                              hipStream_t stream)
{
  (void)in_sizes; (void)n_in; (void)out_size; (void)ws_size;
  const float* x   = (const float*)d_in[0];
  const float* tw1 = (const float*)d_in[1];
  const float* tb1 = (const float*)d_in[2];
  const float* tw2 = (const float*)d_in[3];
  const float* tb2 = (const float*)d_in[4];
  const float* ww1 = (const float*)d_in[5];
  const float* wb1 = (const float*)d_in[6];
  const float* ww2 = (const float*)d_in[7];
  const float* wb2 = (const float*)d_in[8];

  float* zc    = (float*)d_out;              // [8,512,16]
  float* wsout = (float*)d_out + 65536;      // [8,32768,512]

  float* t1      = (float*)d_ws;             // [8,64,8192]
  float* tbuf    = t1   + 4194304;           // [8,64,8192]
  float* w1      = tbuf + 4194304;           // [8,64,8192]
  float* cbuf    = w1   + 4194304;           // [8,512,16]
  float* pcs     = cbuf + 65536;             // [8,64,512]
  float* py      = pcs  + 262144;            // [8,64,512,16]
  float* colsumf = py   + 4194304;           // [8,512]

  // t1 = ReLU(conv(h(x), tw1) + tb1)   M=64, K=768*3
  conv3x1_gemm<<<dim3(64, 1, 8), 256, 0, stream>>>(x,    tw1, tb1, t1,   768, 64, 1, 1);
  // t  = conv(t1, tw2) + tb2           M=64, K=64*3
  conv3x1_gemm<<<dim3(64, 1, 8), 256, 0, stream>>>(t1,   tw2, tb2, tbuf,  64, 64, 0, 0);
  // w1 = ReLU(conv(t, ww1) + wb1)
  conv3x1_gemm<<<dim3(64, 1, 8), 256, 0, stream>>>(tbuf, ww1, wb1, w1,    64, 64, 1, 0);
  // fused conv2 + row-softmax + c = P@zs  (wcf never materialized)
  fused_wcf_softmax_c<<<dim3(128, 8), 128, 0, stream>>>(w1, ww2, wb2, tbuf, cbuf);
  // ws_sm = softmax(-cdist(zs,c))
  dist_softmax_ws<<<dim3(512, 8), 512, 0, stream>>>(tbuf, cbuf, wsout);
  // deterministic column sums + zc
  ws_colsum_part  <<<dim3(64, 8), 512, 0, stream>>>(wsout, tbuf, pcs, py);
  ws_colsum_reduce<<<dim3(8),     512, 0, stream>>>(pcs, py, colsumf, zc);
  // ws = ws_sm / (colsum + 1e-6), in place
  ws_normalize<<<dim3(131072), 256, 0, stream>>>(wsout, colsumf);
}